// LightweightIndustrialDiffusion_33569464385981
// MI455X (gfx1250) — compile-verified
//
#include <hip/hip_runtime.h>
#include <math.h>

typedef __attribute__((ext_vector_type(2))) float v2f;
typedef __attribute__((ext_vector_type(8))) float v8f;

#define N_NODES 2048
#define N_EDGES 65536
#define HID 12
#define HEADS 4
#define TED 16
#define NCLS 4
#define DIN   (NCLS + TED)    // 20
#define HTOT  (HEADS * HID)   // 48

// ---------- helpers ----------
__device__ __forceinline__ int enc_f32(float f) {
  int i = __float_as_int(f);
  return i ^ ((i >> 31) & 0x7fffffff);   // order-preserving float->int map
}
__device__ __forceinline__ float dec_f32(int k) {
  return __int_as_float(k ^ ((k >> 31) & 0x7fffffff));
}

// ---------- init ----------
__global__ void k_fill_f32(float* p, float v, int n) {
  int i = blockIdx.x * blockDim.x + threadIdx.x;
  if (i < n) p[i] = v;
}
__global__ void k_fill_i32(int* p, int v, int n) {
  int i = blockIdx.x * blockDim.x + threadIdx.x;
  if (i < n) p[i] = v;
}

// ---------- time embedding: te = [sin(v),cos(v)] @ time_W + time_b ----------
__global__ void k_time_embed(const int* __restrict__ t_ptr,
                             const float* __restrict__ W,
                             const float* __restrict__ b,
                             float* __restrict__ te) {
  int j = threadIdx.x;
  if (j >= TED) return;
  float t = (float)t_ptr[0] * 0.01f;          // t / T_STEPS
  const float scale = 1.31576288f;            // log(10000)/(TED/2 - 1)
  float acc = b[j];
  for (int i = 0; i < TED; ++i) {
    int d = (i < TED / 2) ? i : i - TED / 2;
    float v = t * expf(-(float)d * scale);
    float e = (i < TED / 2) ? sinf(v) : cosf(v);
    acc += e * W[i * TED + j];
  }
  te[j] = acc;
}

// ---------- xin = concat(x_onehot, broadcast(te)) ----------
__global__ void k_build_xin(const float* __restrict__ x,
                            const float* __restrict__ te,
                            float* __restrict__ xin) {
  int i = blockIdx.x * blockDim.x + threadIdx.x;
  if (i >= N_NODES * DIN) return;
  int n = i / DIN, c = i - n * DIN;
  xin[i] = (c < NCLS) ? x[n * NCLS + c] : te[c - NCLS];
}

// ---------- generic f32 GEMM via V_WMMA_F32_16X16X4_F32 ----------
// Y[M, NCOLS] = X[M, KDIM] @ W[KDIM, NCOLS] + bias
// 4 waves / block, one 16x16 output tile per wave; KDIM/NCOLS compile-time
// so the K-loop fully unrolls with unconditional (64-bit) A loads and
// branch-free masked B loads.
template <int KDIM, int NCOLS>
__global__ void __launch_bounds__(128)
k_gemm_wmma_f32(const float* __restrict__ X,
                const float* __restrict__ W,
                const float* __restrict__ bias,
                float* __restrict__ Y) {
  static_assert(KDIM % 4 == 0, "KDIM must be a multiple of 4");
  int lane = threadIdx.x & 31;
  int wave = threadIdx.x >> 5;
  int m0 = (blockIdx.x * 4 + wave) * 16;
  int n0 = blockIdx.y * 16;
  int mr  = lane & 15;
  int hi  = lane >> 4;            // 0: K+0/K+1 (A) & M+0 (D);  1: K+2/K+3 & M+8
  int row = m0 + mr;
  int col = n0 + mr;
  bool cok   = (col < NCOLS);
  int  ccol  = cok ? col : 0;      // clamped, always-legal address
  float mask = cok ? 1.f : 0.f;
  v8f acc = {0.f, 0.f, 0.f, 0.f, 0.f, 0.f, 0.f, 0.f};
#pragma unroll
  for (int k0 = 0; k0 < KDIM; k0 += 4) {
    int k = k0 + hi * 2;                       // statically in-range
    v2f a, b;
    a.x = X[row * KDIM + k];                   // contiguous pair -> b64 load
    a.y = X[row * KDIM + k + 1];
    b.x = W[k * NCOLS + ccol] * mask;
    b.y = W[(k + 1) * NCOLS + ccol] * mask;
    acc = __builtin_amdgcn_wmma_f32_16x16x4_f32(false, a, false, b,
                                                (short)0, acc, false, false);
  }
  if (cok) {
    float bb = bias ? bias[col] : 0.f;
#pragma unroll
    for (int r = 0; r < 8; ++r) {
      int m = m0 + r + hi * 8;                 // D layout: lanes>=16 hold M+8
      Y[m * NCOLS + col] = acc[r] + bb;
    }
  }
}

// ---------- attention: per-edge score + segment max ----------
__global__ void k_score(const float* __restrict__ q, const float* __restrict__ k,
                        const int* __restrict__ ei, float* __restrict__ score,
                        int* __restrict__ mbits) {
  int idx = blockIdx.x * blockDim.x + threadIdx.x;
  if (idx >= N_EDGES * HEADS) return;
  int e = idx >> 2, h = idx & 3;
  int src = ei[e], dst = ei[N_EDGES + e];
  const float* qp = q + dst * HTOT + h * HID;
  const float* kp = k + src * HTOT + h * HID;
  float s = 0.f;
#pragma unroll
  for (int d = 0; d < HID; ++d) s += qp[d] * kp[d];
  s *= 0.28867513f;                      // 1/sqrt(HID)
  score[idx] = s;
  atomicMax(&mbits[dst * HEADS + h], enc_f32(s));
}

// ---------- decode max; -inf (no in-edges) -> 0 ----------
__global__ void k_mfinal(const int* __restrict__ mbits, float* __restrict__ m) {
  int i = blockIdx.x * blockDim.x + threadIdx.x;
  if (i >= N_NODES * HEADS) return;
  float f = dec_f32(mbits[i]);
  m[i] = (f > -3.0e38f) ? f : 0.f;
}

// ---------- e = exp(score - m[dst]); segment sum ----------
__global__ void k_expsum(const int* __restrict__ ei, const float* __restrict__ m,
                         float* __restrict__ score, float* __restrict__ ssum) {
  int idx = blockIdx.x * blockDim.x + threadIdx.x;
  if (idx >= N_EDGES * HEADS) return;
  int e = idx >> 2, h = idx & 3;
  int dst = ei[N_EDGES + e];
  float ev = expf(score[idx] - m[dst * HEADS + h]);
  score[idx] = ev;
  atomicAdd(&ssum[dst * HEADS + h], ev);
}

// ---------- agg[dst,h,:] += alpha * v[src,h,:] ----------
__global__ void k_aggregate(const int* __restrict__ ei, const float* __restrict__ score,
                            const float* __restrict__ ssum, const float* __restrict__ v,
                            float* __restrict__ agg) {
  int idx = blockIdx.x * blockDim.x + threadIdx.x;
  if (idx >= N_EDGES * HEADS * HID) return;
  int e = idx / (HEADS * HID);
  int r = idx - e * (HEADS * HID);
  int h = r / HID, d = r - h * HID;
  int src = ei[e], dst = ei[N_EDGES + e];
  float alpha = score[e * HEADS + h] / (ssum[dst * HEADS + h] + 1e-16f);
  atomicAdd(&agg[(dst * HEADS + h) * HID + d], alpha * v[src * HTOT + h * HID + d]);
}

// ---------- h_out = relu(mean_heads(agg) + skip) ----------
__global__ void k_combine(const float* __restrict__ agg, const float* __restrict__ skip,
                          float* __restrict__ hout) {
  int i = blockIdx.x * blockDim.x + threadIdx.x;
  if (i >= N_NODES * HID) return;
  int n = i / HID, d = i - n * HID;
  float s = 0.f;
#pragma unroll
  for (int h = 0; h < HEADS; ++h) s += agg[(n * HEADS + h) * HID + d];
  hout[i] = fmaxf(s * 0.25f + skip[i], 0.f);
}

// ---------- dominant kernel: edge logits, bandwidth bound ----------
// out[i,j,c] = relu(A[i]+B[j]) . We2[:,c] + be2[c]
__global__ void __launch_bounds__(256)
k_edge_logits(const float* __restrict__ A, const float* __restrict__ B,
              const float* __restrict__ We2, const float* __restrict__ be2,
              float* __restrict__ out) {
  __shared__ float sA[HID];
  __shared__ float sW[HID * 2];
  __shared__ float sb[2];
  int i = blockIdx.y;
  int tid = threadIdx.x;
  if (tid < HID) sA[tid] = A[i * HID + tid];
  if (tid >= 32 && tid < 32 + HID * 2) sW[tid - 32] = We2[tid - 32];
  if (tid >= 64 && tid < 66) sb[tid - 64] = be2[tid - 64];
  __syncthreads();
  int j = blockIdx.x * 256 + tid;
  const float4* Bp = (const float4*)(B + j * HID);   // 48B rows, 16B aligned
  float4 b0 = Bp[0], b1 = Bp[1], b2 = Bp[2];
  float hb[HID] = {b0.x, b0.y, b0.z, b0.w, b1.x, b1.y, b1.z, b1.w,
                   b2.x, b2.y, b2.z, b2.w};
  float o0 = sb[0], o1 = sb[1];
#pragma unroll
  for (int d = 0; d < HID; ++d) {
    float hv = fmaxf(sA[d] + hb[d], 0.f);
    o0 = fmaf(hv, sW[d * 2 + 0], o0);
    o1 = fmaf(hv, sW[d * 2 + 1], o1);
  }
  v2f res;
  res.x = o0;
  res.y = o1;
  // streaming 64-bit non-temporal store of the 33.5 MB output
  __builtin_nontemporal_store(res, (v2f*)(out + ((size_t)i * N_NODES + j) * 2));
}

// =================================================================
extern "C" void kernel_launch(void* const* d_in, const int* in_sizes, int n_in,
                              void* d_out, int out_size, void* d_ws, size_t ws_size,
                              hipStream_t stream) {
  const float* x      = (const float*)d_in[0];
  const int*   ei     = (const int*)  d_in[1];
  const int*   t_ptr  = (const int*)  d_in[3];
  const float* time_W = (const float*)d_in[4];
  const float* time_b = (const float*)d_in[5];
  const float *Wq1 = (const float*)d_in[6],  *bq1 = (const float*)d_in[7];
  const float *Wk1 = (const float*)d_in[8],  *bk1 = (const float*)d_in[9];
  const float *Wv1 = (const float*)d_in[10], *bv1 = (const float*)d_in[11];
  const float *Ws1 = (const float*)d_in[12], *bs1 = (const float*)d_in[13];
  const float *Wq2 = (const float*)d_in[14], *bq2 = (const float*)d_in[15];
  const float *Wk2 = (const float*)d_in[16], *bk2 = (const float*)d_in[17];
  const float *Wv2 = (const float*)d_in[18], *bv2 = (const float*)d_in[19];
  const float *Ws2 = (const float*)d_in[20], *bs2 = (const float*)d_in[21];
  const float *Wn  = (const float*)d_in[22], *bn  = (const float*)d_in[23];
  const float *We1 = (const float*)d_in[24], *be1 = (const float*)d_in[25];
  const float *We2 = (const float*)d_in[26], *be2 = (const float*)d_in[27];

  float* node_logits = (float*)d_out;                // [2048, 4]
  float* edge_out    = (float*)d_out + N_NODES * NCLS;

  // workspace carve-up (floats)
  float* w = (float*)d_ws;
  float* te    = w;              w += 16;
  float* xin   = w;              w += N_NODES * DIN;
  float* q     = w;              w += N_NODES * HTOT;
  float* kk    = w;              w += N_NODES * HTOT;
  float* vv    = w;              w += N_NODES * HTOT;
  float* skip  = w;              w += N_NODES * HID;
  int*   mbits = (int*)w;        w += N_NODES * HEADS;
  float* m     = w;              w += N_NODES * HEADS;
  float* score = w;              w += N_EDGES * HEADS;
  float* ssum  = w;              w += N_NODES * HEADS;
  float* agg   = w;              w += N_NODES * HEADS * HID;
  float* h1    = w;              w += N_NODES * HID;
  float* h2    = w;              w += N_NODES * HID;
  float* Abuf  = w;              w += N_NODES * HID;
  float* Bbuf  = w;              w += N_NODES * HID;

  const int ENC_NEG_INF = (int)0x807FFFFF;           // enc_f32(-inf)
  const int gx = N_NODES / 64;                       // 4 tiles (waves) per block

  // ---- stage 0: time embedding + xin ----
  k_time_embed<<<1, 32, 0, stream>>>(t_ptr, time_W, time_b, te);
  k_build_xin<<<(N_NODES * DIN + 255) / 256, 256, 0, stream>>>(x, te, xin);

  // ---- layer 1 projections (WMMA f32) ----
  k_gemm_wmma_f32<DIN, HTOT><<<dim3(gx, 3), 128, 0, stream>>>(xin, Wq1, bq1, q);
  k_gemm_wmma_f32<DIN, HTOT><<<dim3(gx, 3), 128, 0, stream>>>(xin, Wk1, bk1, kk);
  k_gemm_wmma_f32<DIN, HTOT><<<dim3(gx, 3), 128, 0, stream>>>(xin, Wv1, bv1, vv);
  k_gemm_wmma_f32<DIN, HID> <<<dim3(gx, 1), 128, 0, stream>>>(xin, Ws1, bs1, skip);

  // ---- layer 1 attention ----
  k_fill_i32<<<(N_NODES * HEADS + 255) / 256, 256, 0, stream>>>(mbits, ENC_NEG_INF, N_NODES * HEADS);
  k_fill_f32<<<(N_NODES * HEADS + 255) / 256, 256, 0, stream>>>(ssum, 0.f, N_NODES * HEADS);
  k_fill_f32<<<(N_NODES * HEADS * HID + 255) / 256, 256, 0, stream>>>(agg, 0.f, N_NODES * HEADS * HID);
  k_score    <<<(N_EDGES * HEADS + 255) / 256, 256, 0, stream>>>(q, kk, ei, score, mbits);
  k_mfinal   <<<(N_NODES * HEADS + 255) / 256, 256, 0, stream>>>(mbits, m);
  k_expsum   <<<(N_EDGES * HEADS + 255) / 256, 256, 0, stream>>>(ei, m, score, ssum);
  k_aggregate<<<(N_EDGES * HEADS * HID + 255) / 256, 256, 0, stream>>>(ei, score, ssum, vv, agg);
  k_combine  <<<(N_NODES * HID + 255) / 256, 256, 0, stream>>>(agg, skip, h1);

  // ---- layer 2 projections ----
  k_gemm_wmma_f32<HID, HTOT><<<dim3(gx, 3), 128, 0, stream>>>(h1, Wq2, bq2, q);
  k_gemm_wmma_f32<HID, HTOT><<<dim3(gx, 3), 128, 0, stream>>>(h1, Wk2, bk2, kk);
  k_gemm_wmma_f32<HID, HTOT><<<dim3(gx, 3), 128, 0, stream>>>(h1, Wv2, bv2, vv);
  k_gemm_wmma_f32<HID, HID> <<<dim3(gx, 1), 128, 0, stream>>>(h1, Ws2, bs2, skip);

  // ---- layer 2 attention ----
  k_fill_i32<<<(N_NODES * HEADS + 255) / 256, 256, 0, stream>>>(mbits, ENC_NEG_INF, N_NODES * HEADS);
  k_fill_f32<<<(N_NODES * HEADS + 255) / 256, 256, 0, stream>>>(ssum, 0.f, N_NODES * HEADS);
  k_fill_f32<<<(N_NODES * HEADS * HID + 255) / 256, 256, 0, stream>>>(agg, 0.f, N_NODES * HEADS * HID);
  k_score    <<<(N_EDGES * HEADS + 255) / 256, 256, 0, stream>>>(q, kk, ei, score, mbits);
  k_mfinal   <<<(N_NODES * HEADS + 255) / 256, 256, 0, stream>>>(mbits, m);
  k_expsum   <<<(N_EDGES * HEADS + 255) / 256, 256, 0, stream>>>(ei, m, score, ssum);
  k_aggregate<<<(N_EDGES * HEADS * HID + 255) / 256, 256, 0, stream>>>(ei, score, ssum, vv, agg);
  k_combine  <<<(N_NODES * HID + 255) / 256, 256, 0, stream>>>(agg, skip, h2);

  // ---- node logits: h2 @ Wn + bn -> d_out[0:8192] ----
  k_gemm_wmma_f32<HID, NCLS><<<dim3(gx, 1), 128, 0, stream>>>(h2, Wn, bn, node_logits);

  // ---- edge MLP factorization: A = h2@We1_top + be1 ; B = h2@We1_bot ----
  k_gemm_wmma_f32<HID, HID><<<dim3(gx, 1), 128, 0, stream>>>(h2, We1,             be1,     Abuf);
  k_gemm_wmma_f32<HID, HID><<<dim3(gx, 1), 128, 0, stream>>>(h2, We1 + HID * HID, nullptr, Bbuf);

  // ---- dominant, bandwidth-bound edge-logits kernel ----
  k_edge_logits<<<dim3(N_NODES / 256, N_NODES), 256, 0, stream>>>(Abuf, Bbuf, We2, be2, edge_out);

  (void)in_sizes; (void)n_in; (void)out_size; (void)ws_size;
}